// EMLTree1D_25134148616782
// MI455X (gfx1250) — compile-verified
//
#include <hip/hip_runtime.h>

// EMLTree1D on gfx1250:
//  - leaf level (the reference's only matmul, cand @ leaf_probs.T) done with
//    V_WMMA_F32_16X16X4_F32 (one real WMMA + one imag WMMA per 16-leaf tile)
//  - each lane owns a depth-3 subtree (8 leaves of one sample) straight from
//    the WMMA D-layout; folds it in registers; __shfl_xor(16) joins halves;
//    a statically-indexed binary-counter ladder folds levels 5..10.
//  - params softmaxed once per block into LDS (32 KB of 320 KB/WGP).

typedef float v2f __attribute__((ext_vector_type(2)));
typedef float v8f __attribute__((ext_vector_type(8)));

#define N_LEAVES          1024
#define N_INTERNAL        1023
#define SAMPLES_PER_WAVE  16
#define SAMPLES_PER_BLOCK 128     // 16 samples * 8 waves
#define N_TILES           64      // 1024 leaves / 16 per WMMA tile

// One tree fold:  out = exp(gateL . mix) - log(gateR . mix), complex,
// with eps-masking of the child term, bypass-to-{1,x} selects, and
// nan_to_num + clip(+-1e30) exactly as the reference.
__device__ __forceinline__ void tree_combine(
    float& outr, float& outi,
    float lr0, float li0, float rr0, float ri0,
    const float* __restrict__ g,   // 6 floats: p0l,p1l,p2l, p0r,p1r,p2r
    float xr, float xi)
{
    const float CHILD_EPS = 2.220446049250313e-16f;
    const float BYPASS    = (float)(1.0 - 2.220446049250313e-16);  // == 1.0f
    const float CLAMP     = 1e30f;

    float p0l = g[0], p1l = g[1], p2l = g[2];
    float p0r = g[3], p1r = g[4], p2r = g[5];

    // masked child contribution (reference zeroes both p2 and child when masked)
    float tlr = (p2l > CHILD_EPS) ? p2l * lr0 : 0.0f;
    float tli = (p2l > CHILD_EPS) ? p2l * li0 : 0.0f;
    float trr = (p2r > CHILD_EPS) ? p2r * rr0 : 0.0f;
    float tri = (p2r > CHILD_EPS) ? p2r * ri0 : 0.0f;

    float lr = p0l + p1l * xr + tlr;
    float li =       p1l * xi + tli;
    float rr = p0r + p1r * xr + trr;
    float ri =       p1r * xi + tri;

    if (p0l > BYPASS) { lr = 1.0f; li = 0.0f; }
    if (p1l > BYPASS) { lr = xr;   li = xi;   }
    if (p0r > BYPASS) { rr = 1.0f; ri = 0.0f; }
    if (p1r > BYPASS) { rr = xr;   ri = xi;   }

    // exp(left) : e^lr * (cos li + i sin li)
    float e = __expf(lr);
    float s, c;
    __sincosf(li, &s, &c);
    float er = e * c;
    float ei = e * s;

    // log(right) : 0.5*log(|r|^2) + i*atan2(ri, rr)
    float mag2 = rr * rr + ri * ri;
    float lgr  = 0.5f * __logf(mag2);
    float lgi  = atan2f(ri, rr);

    float vr = er - lgr;
    float vi = ei - lgi;

    // nan -> 0, then clamp (fminf/fmaxf map +-inf to +-CLAMP)
    vr = (vr != vr) ? 0.0f : fminf(fmaxf(vr, -CLAMP), CLAMP);
    vi = (vi != vi) ? 0.0f : fminf(fmaxf(vi, -CLAMP), CLAMP);
    outr = vr;
    outi = vi;
}

extern "C" __global__ __launch_bounds__(256)
void emltree1d_kernel(const float* __restrict__ x_real,
                      const float* __restrict__ x_imag,
                      const float* __restrict__ leaf_logits,   // (1024,2)
                      const float* __restrict__ gate_logits,   // (1023,2,3)
                      float* __restrict__ out,                 // [2B | 2048 | 6138]
                      int B)
{
    __shared__ float s_leaf[N_LEAVES * 2];      // softmaxed leaf probs, 8 KB
    __shared__ float s_gate[N_INTERNAL * 6];    // softmaxed gate probs, ~24 KB

    const int tid = threadIdx.x;

    // ---- stage softmaxed params into LDS (uniform across samples) ----
    for (int l = tid; l < N_LEAVES; l += blockDim.x) {
        float a = leaf_logits[2 * l + 0], b = leaf_logits[2 * l + 1];
        float m  = fmaxf(a, b);
        float ea = __expf(a - m), eb = __expf(b - m);
        float inv = 1.0f / (ea + eb);
        s_leaf[2 * l + 0] = ea * inv;
        s_leaf[2 * l + 1] = eb * inv;
    }
    for (int n = tid; n < N_INTERNAL * 2; n += blockDim.x) {
        float a = gate_logits[3 * n + 0];
        float b = gate_logits[3 * n + 1];
        float c = gate_logits[3 * n + 2];
        float m  = fmaxf(a, fmaxf(b, c));
        float ea = __expf(a - m), eb = __expf(b - m), ec = __expf(c - m);
        float inv = 1.0f / (ea + eb + ec);
        s_gate[3 * n + 0] = ea * inv;
        s_gate[3 * n + 1] = eb * inv;
        s_gate[3 * n + 2] = ec * inv;
    }
    __syncthreads();

    // tuple outputs 2 & 3: leaf_probs then gate_probs (block 0 only)
    if (blockIdx.x == 0) {
        float* po = out + 2 * B;
        for (int j = tid; j < N_LEAVES * 2; j += blockDim.x)   po[j] = s_leaf[j];
        for (int j = tid; j < N_INTERNAL * 6; j += blockDim.x) po[N_LEAVES * 2 + j] = s_gate[j];
    }

    const int wave = tid >> 5;
    const int lane = tid & 31;
    const int half = lane >> 4;   // which half-wave: selects leaf rows v or v+8
    const int col  = lane & 15;   // sample column inside the wave
    const int sample = blockIdx.x * SAMPLES_PER_BLOCK + wave * SAMPLES_PER_WAVE + col;

    const float xr = x_real[sample];
    const float xi = x_imag[sample];
    const bool  lo = (half == 0);

    // B fragments for V_WMMA_F32_16X16X4_F32 (4x16, K rows striped by half-wave):
    //   real: K0 = ones, K1 = xr  -> level0_re = p0 + p1*xr
    //   imag: K0 = 0,    K1 = xi  -> level0_im =      p1*xi
    v2f bfrag_re, bfrag_im;
    bfrag_re[0] = lo ? 1.0f : 0.0f;   // K=0 (lo) / K=2 (hi, zero)
    bfrag_re[1] = lo ? xr   : 0.0f;   // K=1 (lo) / K=3 (hi, zero)
    bfrag_im[0] = 0.0f;
    bfrag_im[1] = lo ? xi   : 0.0f;

    const v8f czero = {0.f, 0.f, 0.f, 0.f, 0.f, 0.f, 0.f, 0.f};

    // binary-counter fold stack, statically indexed (no v_movrel)
    float s0r = 0.f, s0i = 0.f, s1r = 0.f, s1i = 0.f, s2r = 0.f, s2i = 0.f;
    float s3r = 0.f, s3i = 0.f, s4r = 0.f, s4i = 0.f, s5r = 0.f, s5i = 0.f;
    float s6r = 0.f, s6i = 0.f;

    for (int t = 0; t < N_TILES; ++t) {
        // A fragment: 16 leaves (16t..16t+15) x K = [p0, p1, 0, 0].
        // All lanes load (lane l and l+16 hit the same LDS address -> broadcast),
        // upper half zeroed with cndmask instead of divergent exec masking.
        const v2f av = *(const v2f*)&s_leaf[2 * (16 * t + col)];
        v2f afrag;
        afrag[0] = lo ? av[0] : 0.0f;
        afrag[1] = lo ? av[1] : 0.0f;

        // D layout: lane holds leaves (16t + 8*half + v), v=0..7, for its sample
        v8f dre = __builtin_amdgcn_wmma_f32_16x16x4_f32(
            false, afrag, false, bfrag_re, (short)0, czero, false, false);
        v8f dim = __builtin_amdgcn_wmma_f32_16x16x4_f32(
            false, afrag, false, bfrag_im, (short)0, czero, false, false);

        // ---- in-lane depth-3 subtree fold (7 combines) ----
        float w1r[4], w1i[4];
#pragma unroll
        for (int m = 0; m < 4; ++m) {
            const float* g = &s_gate[6 * (8 * t + 4 * half + m)];          // level-1
            tree_combine(w1r[m], w1i[m],
                         dre[2 * m], dim[2 * m], dre[2 * m + 1], dim[2 * m + 1],
                         g, xr, xi);
        }
        float w2r[2], w2i[2];
#pragma unroll
        for (int m = 0; m < 2; ++m) {
            const float* g = &s_gate[6 * (512 + 4 * t + 2 * half + m)];    // level-2
            tree_combine(w2r[m], w2i[m],
                         w1r[2 * m], w1i[2 * m], w1r[2 * m + 1], w1i[2 * m + 1],
                         g, xr, xi);
        }
        float w3r, w3i;
        {
            const float* g = &s_gate[6 * (768 + 2 * t + half)];            // level-3
            tree_combine(w3r, w3i, w2r[0], w2i[0], w2r[1], w2i[1], g, xr, xi);
        }

        // ---- join the two lane-halves: level-4 node t ----
        float pr = __shfl_xor(w3r, 16);
        float pi = __shfl_xor(w3i, 16);
        float l4lr = half ? pr  : w3r;
        float l4li = half ? pi  : w3i;
        float l4rr = half ? w3r : pr;
        float l4ri = half ? w3i : pi;
        float vr, vi;
        tree_combine(vr, vi, l4lr, l4li, l4rr, l4ri,
                     &s_gate[6 * (896 + t)], xr, xi);

        // ---- binary-counter merge, fully static register indices.
        // Producing level-(5+s) node (t >> (s+1)); gate base = 1024 - 2^(6-s).
        // All branches are wave-uniform (t is uniform) -> s_cbranch only.
        if (t & 1) {
            tree_combine(vr, vi, s0r, s0i, vr, vi, &s_gate[6 * (960 + (t >> 1))], xr, xi);
            if (t & 2) {
                tree_combine(vr, vi, s1r, s1i, vr, vi, &s_gate[6 * (992 + (t >> 2))], xr, xi);
                if (t & 4) {
                    tree_combine(vr, vi, s2r, s2i, vr, vi, &s_gate[6 * (1008 + (t >> 3))], xr, xi);
                    if (t & 8) {
                        tree_combine(vr, vi, s3r, s3i, vr, vi, &s_gate[6 * (1016 + (t >> 4))], xr, xi);
                        if (t & 16) {
                            tree_combine(vr, vi, s4r, s4i, vr, vi, &s_gate[6 * (1020 + (t >> 5))], xr, xi);
                            if (t & 32) {
                                tree_combine(vr, vi, s5r, s5i, vr, vi, &s_gate[6 * 1022], xr, xi);
                                s6r = vr; s6i = vi;
                            } else { s5r = vr; s5i = vi; }
                        } else { s4r = vr; s4i = vi; }
                    } else { s3r = vr; s3i = vi; }
                } else { s2r = vr; s2i = vi; }
            } else { s1r = vr; s1i = vi; }
        } else { s0r = vr; s0i = vi; }
    }

    // after t=63 the 6 trailing merges deposit the root in slot 6
    if (lo) {
        out[2 * sample + 0] = s6r;
        out[2 * sample + 1] = s6i;
    }
}

extern "C" void kernel_launch(void* const* d_in, const int* in_sizes, int n_in,
                              void* d_out, int out_size, void* d_ws, size_t ws_size,
                              hipStream_t stream)
{
    const float* x_real      = (const float*)d_in[0];
    const float* x_imag      = (const float*)d_in[1];
    const float* leaf_logits = (const float*)d_in[2];
    const float* gate_logits = (const float*)d_in[3];
    float*       out         = (float*)d_out;

    const int B = in_sizes[0];                       // 32768
    const int blocks = B / SAMPLES_PER_BLOCK;        // 256
    emltree1d_kernel<<<blocks, 256, 0, stream>>>(
        x_real, x_imag, leaf_logits, gate_logits, out, B);
}